// Sparsemax_48146583388390
// MI455X (gfx1250) — compile-verified
//
#include <hip/hip_runtime.h>

#ifndef __has_builtin
#define __has_builtin(x) 0
#endif

#if defined(__HIP_DEVICE_COMPILE__) && __has_builtin(__builtin_amdgcn_global_load_async_to_lds_b128)
#define USE_ASYNC_LDS 1
#else
#define USE_ASYNC_LDS 0
#endif

#define NEG_BIG (-9999999.9f)
#define INV_ONE_MINUS_TEMP 2.0f   // 1/(1-0.5)

constexpr int ROW = 4096;   // row length (reference shape)
constexpr int T   = 256;    // threads per block = 8 wave32 waves
constexpr int VPT = ROW / T;   // 16 floats per thread
constexpr int F4  = VPT / 4;   // 4 float4 chunks per thread

// Matches the builtin's expected parameter types: int4 (vector_size) pointers
// in AS(1) (global src) and AS(3) (LDS dst).
typedef int gvec4i __attribute__((vector_size(4 * sizeof(int))));
typedef __attribute__((address_space(1))) gvec4i* gptr4i_t;
typedef __attribute__((address_space(3))) gvec4i* lptr4i_t;

// ---------- wave32 butterfly reductions ----------
__device__ __forceinline__ float wave_red_sum(float v) {
#pragma unroll
    for (int off = 16; off > 0; off >>= 1) v += __shfl_xor(v, off, 32);
    return v;
}
__device__ __forceinline__ float wave_red_max(float v) {
#pragma unroll
    for (int off = 16; off > 0; off >>= 1) v = fmaxf(v, __shfl_xor(v, off, 32));
    return v;
}

// ---------- block-wide reduce + broadcast (8 waves) ----------
__device__ __forceinline__ float block_reduce_bcast(float v, float* sh, int tid, bool is_max) {
    v = is_max ? wave_red_max(v) : wave_red_sum(v);
    __syncthreads();                      // guard reuse of sh from earlier call
    if ((tid & 31) == 0) sh[tid >> 5] = v;
    __syncthreads();
    if (tid == 0) {
        float r = sh[0];
#pragma unroll
        for (int w = 1; w < T / 32; ++w) r = is_max ? fmaxf(r, sh[w]) : r + sh[w];
        sh[8] = r;
    }
    __syncthreads();
    return sh[8];
}

__device__ __forceinline__ float2 block_reduce2_bcast(float a, float b, float* sa, float* sb, int tid) {
    a = wave_red_sum(a);
    b = wave_red_sum(b);
    __syncthreads();
    if ((tid & 31) == 0) { sa[tid >> 5] = a; sb[tid >> 5] = b; }
    __syncthreads();
    if (tid == 0) {
        float ra = sa[0], rb = sb[0];
#pragma unroll
        for (int w = 1; w < T / 32; ++w) { ra += sa[w]; rb += sb[w]; }
        sa[8] = ra; sb[8] = rb;
    }
    __syncthreads();
    float2 r; r.x = sa[8]; r.y = sb[8];
    return r;
}

__global__ __launch_bounds__(T)
void sparsemax_rows_kernel(const float* __restrict__ x,
                           const float* __restrict__ mask,
                           float* __restrict__ out) {
    const int tid = threadIdx.x;
    const long long row = blockIdx.x;
    const float* __restrict__ gx = x    + row * (long long)ROW;
    const float* __restrict__ gm = mask + row * (long long)ROW;
    float*       __restrict__ go = out  + row * (long long)ROW;

    __shared__ float sh_a[9];
    __shared__ float sh_b[9];
#if USE_ASYNC_LDS
    __shared__ float xs[ROW];
    __shared__ float ms[ROW];

    // Stage x and mask rows into LDS with gfx1250 async DMA loads (ASYNCcnt).
    // Each lane's B128 lands at its own LDS slot, consumed by the same lane.
#pragma unroll
    for (int j = 0; j < F4; ++j) {
        const int c = j * T + tid;     // float4 index within row
        __builtin_amdgcn_global_load_async_to_lds_b128(
            (gptr4i_t)(gx + c * 4), (lptr4i_t)(xs + c * 4), 0, 0);
        __builtin_amdgcn_global_load_async_to_lds_b128(
            (gptr4i_t)(gm + c * 4), (lptr4i_t)(ms + c * 4), 0, 0);
    }
#if __has_builtin(__builtin_amdgcn_s_wait_asynccnt)
    __builtin_amdgcn_s_wait_asynccnt(0);
#else
    asm volatile("s_wait_asynccnt 0" ::: "memory");
#endif
    __syncthreads();
#endif

    // Build z = ((mask*x + (1-mask)*NEG_BIG) / (1-TEMP)) in registers.
    float z[VPT];
    float lmax = -3.4e38f;
#pragma unroll
    for (int j = 0; j < F4; ++j) {
        const int c = j * T + tid;
#if USE_ASYNC_LDS
        const float4 vx = reinterpret_cast<const float4*>(xs)[c];
        const float4 vm = reinterpret_cast<const float4*>(ms)[c];
#else
        const float4 vx = reinterpret_cast<const float4*>(gx)[c];
        const float4 vm = reinterpret_cast<const float4*>(gm)[c];
#endif
        z[4 * j + 0] = INV_ONE_MINUS_TEMP * fmaf(vm.x, vx.x, (1.0f - vm.x) * NEG_BIG);
        z[4 * j + 1] = INV_ONE_MINUS_TEMP * fmaf(vm.y, vx.y, (1.0f - vm.y) * NEG_BIG);
        z[4 * j + 2] = INV_ONE_MINUS_TEMP * fmaf(vm.z, vx.z, (1.0f - vm.z) * NEG_BIG);
        z[4 * j + 3] = INV_ONE_MINUS_TEMP * fmaf(vm.w, vx.w, (1.0f - vm.w) * NEG_BIG);
        lmax = fmaxf(lmax, fmaxf(fmaxf(z[4 * j + 0], z[4 * j + 1]),
                                 fmaxf(z[4 * j + 2], z[4 * j + 3])));
    }

    // Row max; shift so max(z) == 0 (matches reference normalization).
    const float rmax = block_reduce_bcast(lmax, sh_a, tid, true);
    float lsum = 0.0f;
#pragma unroll
    for (int i = 0; i < VPT; ++i) { z[i] -= rmax; lsum += z[i]; }

    // Michelot fixed-point iteration for the sparsemax threshold tau
    // (exactly the sorted-cumsum tau of the reference, without the sort).
    const float S = block_reduce_bcast(lsum, sh_a, tid, false);
    float tau   = (S - 1.0f) / (float)ROW;
    float prevC = (float)ROW;
#pragma unroll 1
    for (int it = 0; it < 32; ++it) {
        float ss = 0.0f, cc = 0.0f;
#pragma unroll
        for (int i = 0; i < VPT; ++i) {
            if (z[i] > tau) { ss += z[i]; cc += 1.0f; }
        }
        const float2 r = block_reduce2_bcast(ss, cc, sh_a, sh_b, tid);
        if (r.y == prevC) break;          // support stabilized -> tau is exact
        prevC = r.y;
        tau = (r.x - 1.0f) / r.y;
    }

    // out = max(0, z - tau); masked entries are hugely negative -> already 0,
    // so the reference's trailing "* mask" is a no-op here.
#pragma unroll
    for (int j = 0; j < F4; ++j) {
        const int c = j * T + tid;
        float4 o;
        o.x = fmaxf(0.0f, z[4 * j + 0] - tau);
        o.y = fmaxf(0.0f, z[4 * j + 1] - tau);
        o.z = fmaxf(0.0f, z[4 * j + 2] - tau);
        o.w = fmaxf(0.0f, z[4 * j + 3] - tau);
        reinterpret_cast<float4*>(go)[c] = o;
    }
}

extern "C" void kernel_launch(void* const* d_in, const int* in_sizes, int n_in,
                              void* d_out, int out_size, void* d_ws, size_t ws_size,
                              hipStream_t stream) {
    (void)n_in; (void)d_ws; (void)ws_size; (void)out_size;
    const float* x    = (const float*)d_in[0];
    const float* mask = (const float*)d_in[1];
    float* out = (float*)d_out;
    const int rows = in_sizes[0] / ROW;   // 8192 for the reference shape
    sparsemax_rows_kernel<<<rows, T, 0, stream>>>(x, mask, out);
}